// GCNReg_15693810499840
// MI455X (gfx1250) — compile-verified
//
#include <hip/hip_runtime.h>

typedef float v2f __attribute__((ext_vector_type(2)));
typedef float v8f __attribute__((ext_vector_type(8)));

#define DIM 128   // IN_DIM == HIDDEN == 128
#define ODIM 8

// ---------------------------------------------------------------------------
// Init: agg = 0, deg = 1 (self-loop contribution to degree)
// ---------------------------------------------------------------------------
__global__ void gcn_init(float* __restrict__ agg, float* __restrict__ deg, int N) {
    int i = blockIdx.x * blockDim.x + threadIdx.x;
    if (i < N * DIM) agg[i] = 0.0f;
    if (i < N)       deg[i] = 1.0f;
}

// ---------------------------------------------------------------------------
// Degree: deg[dst] += 1 per edge
// ---------------------------------------------------------------------------
__global__ void gcn_degree(const long long* __restrict__ ei, float* __restrict__ deg, int E) {
    int e = blockIdx.x * blockDim.x + threadIdx.x;
    if (e < E) {
        int d = (int)ei[(size_t)E + e];   // edge_index[1] row = targets
        atomicAdd(&deg[d], 1.0f);
    }
}

// ---------------------------------------------------------------------------
// dinv = rsqrt(deg)
// ---------------------------------------------------------------------------
__global__ void gcn_rsqrt(const float* __restrict__ deg, float* __restrict__ dinv, int N) {
    int i = blockIdx.x * blockDim.x + threadIdx.x;
    if (i < N) dinv[i] = rsqrtf(deg[i]);
}

// ---------------------------------------------------------------------------
// xt = x @ W1^T via V_WMMA_F32_16X16X4_F32 (exact fp32 matrix core path).
// Grid: N/16 blocks of 256 threads (8 waves). Block b covers node rows
// [16b, 16b+16); wave w computes the 16x16 output tile at hidden cols
// [16w, 16w+16). K-loop: 128/4 = 32 WMMAs per wave.
//
// ISA layouts (CDNA5 ISA §7.12.2, 32-bit matrices, wave32):
//   A 16x4 : VGPR v, lane group g=lane>>4 -> element (M = lane&15, K = v + 2g)
//   B 4x16 : VGPR v, lane group g         -> element (K = v + 2g, N = lane&15)
//   C/D    : VGPR r, lane group g         -> element (M = r + 8g, N = lane&15)
// So per k-step each lane loads float2 {k+2g, k+2g+1} from its A row (node)
// and from its B row (W1 row hb+n, since B[k][n] = W1[hb+n][k]).
// ---------------------------------------------------------------------------
__global__ __launch_bounds__(256)
void gcn_gemm_xw1(const float* __restrict__ x, const float* __restrict__ W1,
                  float* __restrict__ xt) {
    const int lane = threadIdx.x & 31;
    const int wave = threadIdx.x >> 5;        // hidden tile 0..7
    const int g    = lane >> 4;               // lane group
    const int l15  = lane & 15;
    const int hb   = wave * 16;

    const float* xrow  = x  + (size_t)(blockIdx.x * 16 + l15) * DIM;  // A: M = l15
    const float* w1row = W1 + (size_t)(hb + l15) * DIM;               // B: N = l15

    v8f c = {};
#pragma unroll
    for (int kb = 0; kb < DIM; kb += 4) {
        v2f a = *(const v2f*)(xrow  + kb + 2 * g);
        v2f b = *(const v2f*)(w1row + kb + 2 * g);
        c = __builtin_amdgcn_wmma_f32_16x16x4_f32(
                /*neg_a=*/false, a, /*neg_b=*/false, b,
                /*c_mod=*/(short)0, c, /*reuse_a=*/false, /*reuse_b=*/false);
    }

    float* o = xt + (size_t)(blockIdx.x * 16) * DIM + hb + l15;
#pragma unroll
    for (int r = 0; r < 8; ++r)
        o[(size_t)(r + 8 * g) * DIM] = c[r];    // D: M = r + 8g, N = l15
}

// ---------------------------------------------------------------------------
// Edge scatter: one wave per edge. Lane L handles channels [4L, 4L+4).
// agg[dst] += xt[src] * dinv[src]*dinv[dst]  (global_atomic_add_f32)
// ---------------------------------------------------------------------------
__global__ __launch_bounds__(256)
void gcn_scatter(const long long* __restrict__ ei, const float* __restrict__ xt,
                 const float* __restrict__ dinv, float* __restrict__ agg, int E) {
    int e = blockIdx.x * 8 + (threadIdx.x >> 5);
    if (e >= E) return;
    int lane = threadIdx.x & 31;
    int s = (int)ei[e];
    int d = (int)ei[(size_t)E + e];
    float nrm = dinv[s] * dinv[d];

    float4 v = ((const float4*)(xt + (size_t)s * DIM))[lane];
    float* dp = agg + (size_t)d * DIM + lane * 4;
    atomicAdd(dp + 0, v.x * nrm);
    atomicAdd(dp + 1, v.y * nrm);
    atomicAdd(dp + 2, v.z * nrm);
    atomicAdd(dp + 3, v.w * nrm);
}

// ---------------------------------------------------------------------------
// Head: one wave per node.
//   h = relu(agg + xt*dinv^2 (self-loop, norm = 1/deg) + b1)
//   out = h @ W2^T + b2   (8 outputs, wave32 shuffle reduction)
// ---------------------------------------------------------------------------
__global__ __launch_bounds__(256)
void gcn_head(const float* __restrict__ agg, const float* __restrict__ xt,
              const float* __restrict__ dinv, const float* __restrict__ b1,
              const float* __restrict__ W2, const float* __restrict__ b2,
              float* __restrict__ out, int N) {
    int node = blockIdx.x * 8 + (threadIdx.x >> 5);
    if (node >= N) return;
    int lane = threadIdx.x & 31;

    float di   = dinv[node];
    float self = di * di;              // dinv[i]*dinv[i] = 1/deg[i]

    float4 a  = ((const float4*)(agg + (size_t)node * DIM))[lane];
    float4 xv = ((const float4*)(xt  + (size_t)node * DIM))[lane];
    float4 bb = ((const float4*)b1)[lane];

    float4 h;
    h.x = fmaxf(a.x + xv.x * self + bb.x, 0.0f);
    h.y = fmaxf(a.y + xv.y * self + bb.y, 0.0f);
    h.z = fmaxf(a.z + xv.z * self + bb.z, 0.0f);
    h.w = fmaxf(a.w + xv.w * self + bb.w, 0.0f);

    float acc[ODIM];
#pragma unroll
    for (int o = 0; o < ODIM; ++o) {
        float4 w = ((const float4*)(W2 + (size_t)o * DIM))[lane];
        acc[o] = h.x * w.x + h.y * w.y + h.z * w.z + h.w * w.w;
    }
#pragma unroll
    for (int o = 0; o < ODIM; ++o) {
#pragma unroll
        for (int m = 16; m >= 1; m >>= 1)
            acc[o] += __shfl_xor(acc[o], m, 32);
    }
    if (lane == 0) {
#pragma unroll
        for (int o = 0; o < ODIM; ++o)
            out[(size_t)node * ODIM + o] = acc[o] + b2[o];
    }
}

// ---------------------------------------------------------------------------
extern "C" void kernel_launch(void* const* d_in, const int* in_sizes, int n_in,
                              void* d_out, int out_size, void* d_ws, size_t ws_size,
                              hipStream_t stream) {
    const float*     x  = (const float*)d_in[0];
    const long long* ei = (const long long*)d_in[1];   // int64 edge_index [2, E]
    const float*     W1 = (const float*)d_in[2];       // [128,128]
    const float*     b1 = (const float*)d_in[3];       // [128]
    const float*     W2 = (const float*)d_in[4];       // [8,128]
    const float*     b2 = (const float*)d_in[5];       // [8]
    float*           out = (float*)d_out;              // [N, 8]

    const int N = in_sizes[0] / DIM;                   // 50000
    const int E = in_sizes[1] / 2;                     // 600000

    // Workspace layout: xt[N*128] | agg[N*128] | deg[N] | dinv[N]
    float* xt   = (float*)d_ws;
    float* agg  = xt  + (size_t)N * DIM;
    float* deg  = agg + (size_t)N * DIM;
    float* dinv = deg + N;

    gcn_init  <<<(N * DIM + 255) / 256, 256, 0, stream>>>(agg, deg, N);
    gcn_degree<<<(E + 255) / 256,       256, 0, stream>>>(ei, deg, E);
    gcn_rsqrt <<<(N + 255) / 256,       256, 0, stream>>>(deg, dinv, N);
    gcn_gemm_xw1<<<N / 16,              256, 0, stream>>>(x, W1, xt);
    gcn_scatter <<<(E + 7) / 8,         256, 0, stream>>>(ei, xt, dinv, agg, E);
    gcn_head    <<<(N + 7) / 8,         256, 0, stream>>>(agg, xt, dinv, b1, W2, b2, out, N);
}